// SSDLoss_5746666242703
// MI455X (gfx1250) — compile-verified
//
#include <hip/hip_runtime.h>
#include <stdint.h>

#define B_  64
#define A_  8732
#define C_  81
#define EPSF 1e-7f

typedef uint32_t u32x4 __attribute__((ext_vector_type(4)));
typedef uint32_t u32x8 __attribute__((ext_vector_type(8)));
typedef float    f32x4 __attribute__((ext_vector_type(4)));

__device__ __forceinline__ uint32_t rfl(uint32_t x) {
  return __builtin_amdgcn_readfirstlane(x);
}

// ---------------------------------------------------------------------------
// Pass 1: 128 threads = 4 independent wave32s. Each wave TDM-stages 32
// anchors x 81 classes (cnt*81 contiguous f32) of the confidence tensor into
// its private LDS slab via tensor_load_to_lds (non-temporal: 181 MB one-shot
// stream must not evict the L2-resident pass-2 working set), waits on
// TENSORcnt, then each lane computes log-sum-exp / CE / neg_ce / GIoU for one
// anchor. Per-wave partials written deterministically (no atomics/barriers).
// ---------------------------------------------------------------------------
__global__ void __launch_bounds__(128) ssd_pass1(
    const float* __restrict__ conf, const float* __restrict__ loc,
    const float* __restrict__ tgt, float* __restrict__ negv,
    float* __restrict__ wp_pos, float* __restrict__ wp_loc,
    float* __restrict__ wp_nm, int nblkx)
{
  __shared__ float sh[4 * 32 * C_];                 // 41,472 B
  const int lane = threadIdx.x & 31;
  const int wave = threadIdx.x >> 5;
  const int row  = blockIdx.y;
  const int a0   = blockIdx.x * 128 + wave * 32;
  const int wp   = row * (nblkx * 4) + blockIdx.x * 4 + wave;

  int cnt = A_ - a0;
  cnt = cnt < 0 ? 0 : (cnt > 32 ? 32 : cnt);
  if (cnt == 0) {                                   // wave-uniform branch
    if (lane == 0) { wp_pos[wp] = 0.f; wp_loc[wp] = 0.f; wp_nm[wp] = 0.f; }
    return;
  }

  float* slab = &sh[wave * 32 * C_];

  // ---- TDM descriptor (D#): one-row tile of n = cnt*81 f32 elements ----
  const uint64_t gaddr = (uint64_t)(uintptr_t)(conf + ((size_t)row * A_ + a0) * C_);
  const uint32_t ldsa  = (uint32_t)(uintptr_t)slab;   // low 32b of flat = LDS offset
  const uint32_t n     = (uint32_t)cnt * (uint32_t)C_; // <= 2592, fits tile_dim0

  u32x4 g0;
  g0[0] = rfl(1u);                                   // count=1 (valid user D#)
  g0[1] = rfl(ldsa);                                 // lds_addr
  g0[2] = rfl((uint32_t)gaddr);                      // global_addr[31:0]
  g0[3] = rfl((((uint32_t)(gaddr >> 32)) & 0x01FFFFFFu) | (2u << 30)); // [56:32], type=2

  u32x8 g1;
  g1[0] = rfl(2u << 16);                             // data_size = 4 bytes
  g1[1] = rfl((n & 0xFFFFu) << 16);                  // tensor_dim0[15:0]
  g1[2] = rfl(((n >> 16) & 0xFFFFu) | (1u << 16));   // tensor_dim0[31:16], tensor_dim1=1
  g1[3] = rfl((n & 0xFFFFu) << 16);                  // tile_dim0 = n
  g1[4] = rfl(1u);                                   // tile_dim1=1, tile_dim2=0
  g1[5] = rfl(n);                                    // tensor_dim0_stride = n
  g1[6] = rfl(0u);
  g1[7] = rfl(0u);

  asm volatile("tensor_load_to_lds %0, %1 th:TH_LOAD_NT" :: "s"(g0), "s"(g1) : "memory");
  __builtin_amdgcn_s_wait_tensorcnt(0);
  asm volatile("" ::: "memory");

  const bool active = lane < cnt;
  const int  aIdx   = a0 + lane;
  const float* my   = &slab[lane * C_];

  // streaming log-sum-exp over 81 classes (LDS-resident)
  float m = -3.4e38f;
  for (int j = 0; j < C_; ++j) m = fmaxf(m, my[j]);
  float s = 0.f;
  for (int j = 0; j < C_; ++j) s += expf(my[j] - m);
  const float lse = m + logf(s);

  int label = 0;
  f32x4 p = (f32x4)(0.f);
  float tx1 = 0.f, ty1 = 0.f, tx2 = 0.f, ty2 = 0.f;
  if (active) {
    const float* tg = tgt + ((size_t)row * A_ + aIdx) * 5;
    tx1 = __builtin_nontemporal_load(tg + 0);
    ty1 = __builtin_nontemporal_load(tg + 1);
    tx2 = __builtin_nontemporal_load(tg + 2);
    ty2 = __builtin_nontemporal_load(tg + 3);
    label = (int)__builtin_nontemporal_load(tg + 4);
    p = __builtin_nontemporal_load(((const f32x4*)loc) + ((size_t)row * A_ + aIdx));
  }
  const bool pos = active && (label > 0);
  const float ce = lse - my[label];
  float negce = fmaxf(lse - my[0], 0.f);             // >= 0 mathematically
  negce = pos ? 0.f : negce;                         // positives -> 0 (sum-preserving)
  if (active) negv[(size_t)row * A_ + aIdx] = negce; // stays L2-resident for pass 2

  // GIoU loss
  float area_p = (p.z - p.x) * (p.w - p.y);
  float area_t = (tx2 - tx1) * (ty2 - ty1);
  float iw = fmaxf(fminf(p.z, tx2) - fmaxf(p.x, tx1), 0.f);
  float ih = fmaxf(fminf(p.w, ty2) - fmaxf(p.y, ty1), 0.f);
  float inter = iw * ih;
  float uni   = area_p + area_t - inter;
  float iou   = inter / (uni + EPSF);
  float cw2   = fmaxf(p.z, tx2) - fminf(p.x, tx1);
  float ch2   = fmaxf(p.w, ty2) - fminf(p.y, ty1);
  float area_c = cw2 * ch2;
  float gl = 1.f - (iou - (area_c - uni) / (area_c + EPSF));

  float vp = pos ? ce : 0.f;
  float vl = pos ? gl : 0.f;
  float vn = pos ? 1.f : 0.f;
  for (int o = 16; o > 0; o >>= 1) {                 // wave32 reduction
    vp += __shfl_xor(vp, o, 32);
    vl += __shfl_xor(vl, o, 32);
    vn += __shfl_xor(vn, o, 32);
  }
  if (lane == 0) { wp_pos[wp] = vp; wp_loc[wp] = vl; wp_nm[wp] = vn; }
}

// ---------------------------------------------------------------------------
// Pass 2: one 256-thread block per batch row. Deterministic tree-reduction of
// wave partials, then exact top-k sum of neg_ce via 4-round radix select
// (all keys >= 0, so f32 bits order as unsigned).
// ---------------------------------------------------------------------------
__device__ __forceinline__ float blockReduce256(float v, float* red) {
  const int t = threadIdx.x;
  red[t] = v; __syncthreads();
  for (int o = 128; o > 0; o >>= 1) {
    if (t < o) red[t] += red[t + o];
    __syncthreads();
  }
  float r = red[0]; __syncthreads();
  return r;
}

__global__ void __launch_bounds__(256) ssd_pass2(
    const float* __restrict__ negv, const float* __restrict__ wp_pos,
    const float* __restrict__ wp_loc, const float* __restrict__ wp_nm,
    float* __restrict__ rowres, int wpPerRow)
{
  __shared__ float red[256];
  __shared__ unsigned hist[256];
  __shared__ unsigned s_prefix;
  __shared__ int s_krem;
  __shared__ float s_p, s_l, s_n;

  const int t = threadIdx.x;
  const int row = blockIdx.x;

  float p = 0.f, l = 0.f, nmf = 0.f;
  for (int i = t; i < wpPerRow; i += 256) {
    p   += wp_pos[row * wpPerRow + i];
    l   += wp_loc[row * wpPerRow + i];
    nmf += wp_nm [row * wpPerRow + i];
  }
  p   = blockReduce256(p, red);
  l   = blockReduce256(l, red);
  nmf = blockReduce256(nmf, red);
  if (t == 0) { s_p = p; s_l = l; s_n = nmf; }
  __syncthreads();

  const int nm = (int)(s_n + 0.5f);
  int k = 3 * nm; if (k > A_) k = A_;

  float neg_loss = 0.f;
  if (k > 0) {
    const unsigned* keys = (const unsigned*)(negv + (size_t)row * A_);
    unsigned prefix = 0u;
    int krem = k;
    for (int round = 0; round < 4; ++round) {
      const int shift = 24 - 8 * round;
      hist[t] = 0u; __syncthreads();
      const unsigned pmask = (round == 0) ? 0u : (0xFFFFFFFFu << (shift + 8));
      for (int i = t; i < A_; i += 256) {
        unsigned b = keys[i];
        if ((b & pmask) == prefix) atomicAdd(&hist[(b >> shift) & 0xFFu], 1u);
      }
      __syncthreads();
      if (t == 0) {
        int cum = 0, bin = 0;
        for (int bb = 255; bb >= 0; --bb) {
          int c = (int)hist[bb];
          if (cum + c >= krem) { bin = bb; s_krem = krem - cum; break; }
          cum += c;
        }
        s_prefix = prefix | ((unsigned)bin << shift);
      }
      __syncthreads();
      prefix = s_prefix; krem = s_krem;
    }
    // prefix == bit pattern of the k-th largest value T; krem = copies of T kept
    const float T = __uint_as_float(prefix);
    float sum = 0.f;
    for (int i = t; i < A_; i += 256) {
      unsigned b = keys[i];
      if (b > prefix) sum += __uint_as_float(b);
    }
    sum = blockReduce256(sum, red);
    if (t == 0) neg_loss = sum + (float)krem * T;
  }

  if (t == 0) {
    rowres[row * 4 + 0] = s_p;
    rowres[row * 4 + 1] = s_l;
    rowres[row * 4 + 2] = (float)nm;
    rowres[row * 4 + 3] = neg_loss;
  }
}

// ---------------------------------------------------------------------------
// Pass 3: finalize the 3 scalars.
// ---------------------------------------------------------------------------
__global__ void __launch_bounds__(64) ssd_pass3(const float* __restrict__ rowres,
                                               float* __restrict__ out)
{
  __shared__ float sp[64], sl[64], sn[64], sg[64];
  const int t = threadIdx.x;
  sp[t] = rowres[t * 4 + 0];
  sl[t] = rowres[t * 4 + 1];
  sn[t] = rowres[t * 4 + 2];
  sg[t] = rowres[t * 4 + 3];
  __syncthreads();
  for (int o = 32; o > 0; o >>= 1) {
    if (t < o) { sp[t] += sp[t+o]; sl[t] += sl[t+o]; sn[t] += sn[t+o]; sg[t] += sg[t+o]; }
    __syncthreads();
  }
  if (t == 0) {
    const float posL = sp[0], locL = sl[0], hits = sn[0], negL = sg[0];
    const float confL = posL + negL;
    const float total = confL + locL;          // ALPHA = 1
    const bool  has   = hits > 0.5f;
    const float denom = fmaxf(hits, 1.f);
    out[0] = has ? total / denom : 0.f;
    out[1] = has ? confL / denom : 0.f;
    out[2] = has ? locL  / denom : 0.f;
  }
}

extern "C" void kernel_launch(void* const* d_in, const int* in_sizes, int n_in,
                              void* d_out, int out_size, void* d_ws, size_t ws_size,
                              hipStream_t stream) {
  (void)in_sizes; (void)n_in; (void)out_size; (void)ws_size;
  const float* conf = (const float*)d_in[0];   // (64, 8732, 81)
  const float* loc  = (const float*)d_in[1];   // (64, 8732, 4)
  const float* tgt  = (const float*)d_in[2];   // (64, 8732, 5)
  float* out = (float*)d_out;                  // 3 scalars
  float* ws  = (float*)d_ws;

  const int nblkx    = (A_ + 127) / 128;       // 69
  const int wpPerRow = nblkx * 4;              // 276 wave partials per row

  float* negv   = ws;                               // B*A
  float* wp_pos = negv   + (size_t)B_ * A_;
  float* wp_loc = wp_pos + (size_t)B_ * wpPerRow;
  float* wp_nm  = wp_loc + (size_t)B_ * wpPerRow;
  float* rowres = wp_nm  + (size_t)B_ * wpPerRow;   // B*4

  dim3 g1(nblkx, B_);
  ssd_pass1<<<g1, 128, 0, stream>>>(conf, loc, tgt, negv, wp_pos, wp_loc, wp_nm, nblkx);
  ssd_pass2<<<B_, 256, 0, stream>>>(negv, wp_pos, wp_loc, wp_nm, rowres, wpPerRow);
  ssd_pass3<<<1, 64, 0, stream>>>(rowres, out);
}